// CNN_CRF_36258113913085
// MI455X (gfx1250) — compile-verified
//
#include <hip/hip_runtime.h>

// ---------------------------------------------------------------------------
// CRF Viterbi decode (max-plus recurrence) for MI455X / gfx1250.
// B=4096, T=700, K=10. Memory-bound (~175 MB total traffic, ~8us roofline at
// 23.3 TB/s). CDNA5-specific paths used: Tensor Data Mover
// (tensor_load_to_lds) for global->LDS staging of feats, s_wait_tensorcnt,
// global_prefetch. WMMA is intentionally NOT used: Viterbi is a max-plus
// semiring product, not expressible as multiply-accumulate.
// Outputs (concatenated float32): path_scores [B], best_paths [B,T] as floats.
// ---------------------------------------------------------------------------

#define KTAG 10
#define START_TAG 8
#define STOP_TAG 9
#define NEGV (-10000.0f)
#define BLK 256
#define CT 5     // timesteps per TDM chunk (forward)
#define CT2 70   // timesteps per transpose tile (backtrace)

typedef unsigned int vu4 __attribute__((ext_vector_type(4)));
typedef int          vi8 __attribute__((ext_vector_type(8)));
typedef int          vi4 __attribute__((ext_vector_type(4)));

// Issue a TDM 2-D tile load: `rows` rows of `tileX` 4-byte elements,
// row stride `strideX` elements, from global `gaddr` into LDS byte offset
// `lds_off`. Descriptor layout per CDNA5 ISA section 8 (D# groups 0/1).
__device__ __forceinline__ void tdm_load_tile_f32(unsigned long long gaddr,
                                                  unsigned int lds_off,
                                                  unsigned int tileX,
                                                  unsigned int rows,
                                                  unsigned int strideX,
                                                  unsigned int dim0,
                                                  unsigned int dim1) {
  vu4 g0;
  g0.x = 1u;                                              // count=1, user mode
  g0.y = lds_off;                                         // lds_addr [63:32]
  g0.z = (unsigned int)gaddr;                             // global_addr lo
  g0.w = (unsigned int)((gaddr >> 32) & 0x1FFFFFFull)     // global_addr hi
         | (2u << 30);                                    // type = 2 (image)
  vi8 g1;
  g1[0] = (int)(2u << 16);                                // data_size=2 -> 4B
  g1[1] = (int)((dim0 & 0xFFFFu) << 16);                  // tensor_dim0 lo16
  g1[2] = (int)((dim0 >> 16) | ((dim1 & 0xFFFFu) << 16)); // dim0 hi / dim1 lo
  g1[3] = (int)((dim1 >> 16) | (tileX << 16));            // dim1 hi / tile_dim0
  g1[4] = (int)rows;                                      // tile_dim1 (dim2=0)
  g1[5] = (int)strideX;                                   // tensor_dim0_stride lo32
  g1[6] = 0;                                              // stride hi / stride1 lo
  g1[7] = 0;
  vi4 z4 = {0, 0, 0, 0};
#if __clang_major__ >= 23
  vi8 z8 = {0, 0, 0, 0, 0, 0, 0, 0};
  __builtin_amdgcn_tensor_load_to_lds(g0, g1, z4, z4, z8, 0);
#else
  __builtin_amdgcn_tensor_load_to_lds(g0, g1, z4, z4, 0);
#endif
}

// ---------------------------------------------------------------------------
// Kernel 1: Viterbi forward pass.
// One thread per batch element. Transitions live in SGPRs/registers (uniform,
// constant-indexed after full unroll). feats staged via TDM into a
// double-buffered LDS tile [BLK][CT*KTAG]. Backpointers nibble-packed
// (10 tags x 4 bits) into u64, stored [T][B] (coalesced). Final forward
// variables stored [K][B] (coalesced).
// ---------------------------------------------------------------------------
__global__ void crf_viterbi_forward(const float* __restrict__ feats,
                                    const int* __restrict__ lengths,
                                    const float* __restrict__ trans,
                                    unsigned long long* __restrict__ bp,
                                    float* __restrict__ fv_out,
                                    int B, int T) {
  extern __shared__ float smem[];          // [2][BLK][CT*KTAG]
  const int tid = threadIdx.x;
  const int b0  = blockIdx.x * BLK;
  const int b   = b0 + tid;
  const int len = lengths[b];

  // Transitions -> registers (uniform across lanes, constant-indexed).
  float tr[KTAG * KTAG];
#pragma unroll
  for (int i = 0; i < KTAG * KTAG; ++i) tr[i] = trans[i];

  float fv[KTAG];
#pragma unroll
  for (int k = 0; k < KTAG; ++k) fv[k] = NEGV;
  fv[START_TAG] = 0.0f;

  const int NC = (T + CT - 1) / CT;
  const unsigned int bufBytes = (unsigned int)(BLK * CT * KTAG * sizeof(float));
  const unsigned int ldsBase  = (unsigned int)(size_t)(void*)smem; // LDS offset
  const unsigned long long gBase =
      (unsigned long long)(size_t)feats +
      4ull * (unsigned long long)b0 * (unsigned long long)T * KTAG;

  // Prologue: stage chunk 0.
  if (tid < 32) {
    tdm_load_tile_f32(gBase, ldsBase, CT * KTAG, BLK,
                      (unsigned int)(T * KTAG),
                      (unsigned int)(T * KTAG), (unsigned int)B);
  }

  for (int c = 0; c < NC; ++c) {
    if (tid < 32) {
      if (c + 1 < NC) {
        // Prefetch next chunk into the other buffer, then wait until the
        // current chunk's DMA (in-order per wave) has landed.
        tdm_load_tile_f32(gBase + 4ull * (unsigned long long)((c + 1) * CT) * KTAG,
                          ldsBase + ((unsigned int)(c + 1) & 1u) * bufBytes,
                          CT * KTAG, BLK, (unsigned int)(T * KTAG),
                          (unsigned int)(T * KTAG), (unsigned int)B);
        __builtin_amdgcn_s_wait_tensorcnt(1);
      } else {
        __builtin_amdgcn_s_wait_tensorcnt(0);
      }
    }
    __syncthreads();  // chunk c visible to all waves

    const float* fb = smem + (c & 1) * (BLK * CT * KTAG) + tid * (CT * KTAG);
#pragma unroll
    for (int tl = 0; tl < CT; ++tl) {
      const int t = c * CT + tl;
      const bool act = (t < len) && (t < T);
      float f[KTAG];
#pragma unroll
      for (int k = 0; k < KTAG; ++k) f[k] = fb[tl * KTAG + k];

      unsigned long long w = 0ull;
      float nv[KTAG];
#pragma unroll
      for (int nx = 0; nx < KTAG; ++nx) {
        float m = fv[0] + tr[nx * KTAG + 0];
        int id = 0;
#pragma unroll
        for (int pv = 1; pv < KTAG; ++pv) {
          const float s = fv[pv] + tr[nx * KTAG + pv];
          if (s > m) { m = s; id = pv; }   // strict > : first-max == jnp.argmax
        }
        nv[nx] = m + f[nx];
        w |= (unsigned long long)id << (4 * nx);
      }
      if (act) {
#pragma unroll
        for (int k = 0; k < KTAG; ++k) fv[k] = nv[k];
        bp[(size_t)t * B + b] = w;         // [T][B]: coalesced store
      }
    }
    __syncthreads();  // all consumers done before buffer (c&1) is reused
  }

#pragma unroll
  for (int k = 0; k < KTAG; ++k) fv_out[(size_t)k * B + b] = fv[k];  // [K][B]
}

// ---------------------------------------------------------------------------
// Kernel 2: terminal scores + backtrace.
// Reverse scan over t; bp reads are coalesced across lanes ([T][B]).
// Path (tag ids as float32) written through an LDS transpose tile so [B][T]
// stores go out as contiguous 280-byte row segments.
// ---------------------------------------------------------------------------
__global__ void crf_viterbi_backtrace(const float* __restrict__ trans,
                                      const int* __restrict__ lengths,
                                      const unsigned long long* __restrict__ bp,
                                      const float* __restrict__ fv_in,
                                      float* __restrict__ scores,
                                      float* __restrict__ path,
                                      int B, int T) {
  extern __shared__ float tile[];          // [BLK][CT2]
  const int tid = threadIdx.x;
  const int b0  = blockIdx.x * BLK;
  const int b   = b0 + tid;
  const int len = lengths[b];

  // terminal = fv + transitions[STOP_TAG, :]; max / first-argmax
  float m = fv_in[b] + trans[STOP_TAG * KTAG + 0];
  int tag = 0;
#pragma unroll
  for (int k = 1; k < KTAG; ++k) {
    const float s = fv_in[(size_t)k * B + b] + trans[STOP_TAG * KTAG + k];
    if (s > m) { m = s; tag = k; }
  }
  scores[b] = m;

  const int NC2 = (T + CT2 - 1) / CT2;
  for (int c = NC2 - 1; c >= 0; --c) {
    const int t0 = c * CT2;
    // Hint the next (descending-address) bp chunk: global_prefetch_b8.
    if (t0 >= CT2)
      __builtin_prefetch(&bp[(size_t)(t0 - CT2) * B + b], 0, 1);

    for (int tt = CT2 - 1; tt >= 0; --tt) {
      const int t = t0 + tt;
      float outv = 0.0f;
      if (t < T) {
        const bool act = (t < len);
        outv = act ? (float)tag : 0.0f;             // zero padding past length
        const unsigned long long w = bp[(size_t)t * B + b];
        const int p = (int)((w >> (4 * tag)) & 0xFull);
        if (act) tag = p;                           // frozen past length
      }
      tile[tid * CT2 + tt] = outv;
    }
    __syncthreads();
    // Cooperative transposed store: consecutive lanes write consecutive t.
#pragma unroll 1
    for (int j = 0; j < CT2; ++j) {
      const int idx = j * BLK + tid;
      const int row = idx / CT2;
      const int col = idx - row * CT2;
      const int t = t0 + col;
      if (t < T) path[(size_t)(b0 + row) * T + t] = tile[row * CT2 + col];
    }
    __syncthreads();
  }
}

// ---------------------------------------------------------------------------
extern "C" void kernel_launch(void* const* d_in, const int* in_sizes, int n_in,
                              void* d_out, int out_size, void* d_ws, size_t ws_size,
                              hipStream_t stream) {
  const float* feats   = (const float*)d_in[0];   // [B, T, K] f32
  const int*   lengths = (const int*)d_in[1];     // [B] i32
  const float* trans   = (const float*)d_in[2];   // [K, K] f32

  const int B = in_sizes[1];
  const int T = in_sizes[0] / (B * KTAG);

  float* scores = (float*)d_out;                  // [B] f32
  float* path   = scores + B;                     // [B, T] tag ids as f32

  unsigned long long* bp = (unsigned long long*)d_ws;      // [T][B] u64
  float* fv = (float*)(bp + (size_t)T * B);                // [K][B] f32

  const int blocks = B / BLK;                               // B=4096 -> 16

  const size_t sh1 = (size_t)2 * BLK * CT * KTAG * sizeof(float);   // 100 KB
  crf_viterbi_forward<<<blocks, BLK, sh1, stream>>>(feats, lengths, trans,
                                                    bp, fv, B, T);

  const size_t sh2 = (size_t)BLK * CT2 * sizeof(float);             // 70 KB
  crf_viterbi_backtrace<<<blocks, BLK, sh2, stream>>>(trans, lengths, bp, fv,
                                                      scores, path, B, T);
}